// PinocchioPandaDynamics_48808008351849
// MI455X (gfx1250) — compile-verified
//
#include <hip/hip_runtime.h>
#include <math.h>

#define NJ 7
#define DT_STEP 0.01f
#define G_ACC 9.81f

// ---------------------------------------------------------------------------
// CDNA5 Tensor Data Mover staging (guarded: falls back to plain LDS staging)
// ---------------------------------------------------------------------------
#if defined(__HIP_DEVICE_COMPILE__) && defined(__gfx1250__) && \
    __has_builtin(__builtin_amdgcn_tensor_load_to_lds) &&      \
    __has_builtin(__builtin_amdgcn_s_wait_tensorcnt)
#define USE_TDM 1
#else
#define USE_TDM 0
#endif

#if USE_TDM
typedef unsigned int v4u_ __attribute__((ext_vector_type(4)));
typedef int v8i_ __attribute__((ext_vector_type(8)));
typedef int v4i_ __attribute__((ext_vector_type(4)));

__device__ __forceinline__ unsigned __lds_byte_off(void* p) {
  // generic -> LDS address space cast, then take the 32-bit LDS offset
  return (unsigned)(unsigned long long)(__attribute__((address_space(3))) char*)p;
}

// 1-row tensor_load_to_lds: copy n contiguous f32 from global to LDS.
// D# layout per CDNA5 ISA (group0 + group1; groups 2/3 zero).
__device__ __forceinline__ void tdm_load_f32(const float* gsrc, void* ldst, unsigned n) {
  unsigned long long ga = (unsigned long long)gsrc;
  v4u_ g0;
  g0.x = 1u;                                                // count=1, user mode
  g0.y = __lds_byte_off(ldst);                              // lds_addr (bytes)
  g0.z = (unsigned)ga;                                      // global_addr[31:0]
  g0.w = (unsigned)((ga >> 32) & 0x01FFFFFFu) | (2u << 30); // addr[56:32] | type=2
  v8i_ g1;
  g1[0] = (int)(2u << 16);                                  // data_size = 4 bytes
  g1[1] = (int)((n & 0xFFFFu) << 16);                       // tensor_dim0[15:0]
  g1[2] = (int)(((n >> 16) & 0xFFFFu) | (1u << 16));        // dim0 hi | tensor_dim1=1
  g1[3] = (int)((n & 0xFFFFu) << 16);                       // tile_dim0 = n
  g1[4] = 1;                                                // tile_dim1=1, tile_dim2=0
  g1[5] = (int)n;                                           // tensor_dim0_stride
  g1[6] = 0;
  g1[7] = 0;
  v4i_ z4 = {0, 0, 0, 0};
#if __clang_major__ >= 23
  v8i_ z8 = {0, 0, 0, 0, 0, 0, 0, 0};
  __builtin_amdgcn_tensor_load_to_lds(g0, g1, z4, z4, z8, 0);
#else
  __builtin_amdgcn_tensor_load_to_lds(g0, g1, z4, z4, 0);
#endif
}
#endif  // USE_TDM

// ---------------------------------------------------------------------------
// small matrix helpers (3x3 row-major)
// ---------------------------------------------------------------------------
__device__ __forceinline__ void m3mul(const float* A, const float* B, float* C) {
#pragma unroll
  for (int r = 0; r < 3; ++r)
#pragma unroll
    for (int c = 0; c < 3; ++c)
      C[3 * r + c] = A[3 * r + 0] * B[c] + A[3 * r + 1] * B[3 + c] + A[3 * r + 2] * B[6 + c];
}
__device__ __forceinline__ void m3mul_acc(const float* A, const float* B, float* C) {
#pragma unroll
  for (int r = 0; r < 3; ++r)
#pragma unroll
    for (int c = 0; c < 3; ++c)
      C[3 * r + c] += A[3 * r + 0] * B[c] + A[3 * r + 1] * B[3 + c] + A[3 * r + 2] * B[6 + c];
}
__device__ __forceinline__ void m3Tmul(const float* A, const float* B, float* C) {  // C = A^T B
#pragma unroll
  for (int r = 0; r < 3; ++r)
#pragma unroll
    for (int c = 0; c < 3; ++c)
      C[3 * r + c] = A[r] * B[c] + A[3 + r] * B[3 + c] + A[6 + r] * B[6 + c];
}
__device__ __forceinline__ void m3Tmul_acc(const float* A, const float* B, float* C) {
#pragma unroll
  for (int r = 0; r < 3; ++r)
#pragma unroll
    for (int c = 0; c < 3; ++c)
      C[3 * r + c] += A[r] * B[c] + A[3 + r] * B[3 + c] + A[6 + r] * B[6 + c];
}
__device__ __forceinline__ void m3Tvec(const float* A, const float* v, float* o) {  // o = A^T v
#pragma unroll
  for (int r = 0; r < 3; ++r) o[r] = A[r] * v[0] + A[3 + r] * v[1] + A[6 + r] * v[2];
}

// ---------------------------------------------------------------------------
// main kernel: one thread per batch element, all 3 ABA sweeps fully unrolled
// ---------------------------------------------------------------------------
__global__ __launch_bounds__(256) void panda_aba_kernel(
    const float* __restrict__ x, const float* __restrict__ u,
    const float* __restrict__ axes, const float* __restrict__ Rfix,
    const float* __restrict__ pfix, const float* __restrict__ mass,
    const float* __restrict__ com, const float* __restrict__ inertia,
    float* __restrict__ out, int B) {
  __shared__ float sAxes[NJ * 3];
  __shared__ float sR[NJ * 9];
  __shared__ float sP[NJ * 3];
  __shared__ float sI6[NJ * 36];

  const int t = threadIdx.x;
  const long long b = (long long)blockIdx.x * blockDim.x + t;

  if (b < B) {  // pull this thread's rows into the WGP-level caches while we stage
    __builtin_prefetch(&x[b * 14], 0, 3);
    __builtin_prefetch(&u[b * NJ], 0, 3);
  }

#if USE_TDM
  if (t == 0) {
    tdm_load_f32(axes, sAxes, NJ * 3);
    tdm_load_f32(Rfix, sR, NJ * 9);
    tdm_load_f32(pfix, sP, NJ * 3);
    __builtin_amdgcn_s_wait_tensorcnt(0);
  }
#else
  if (t < NJ * 3) sAxes[t] = axes[t];
  if (t < NJ * 9) sR[t] = Rfix[t];
  if (t < NJ * 3) sP[t] = pfix[t];
#endif

  // spatial inertia I6[i] (6x6) built by 7 threads
  if (t < NJ) {
    const float m = mass[t];
    const float c0 = com[3 * t + 0], c1 = com[3 * t + 1], c2 = com[3 * t + 2];
    const float C[9] = {0.f, -c2, c1, c2, 0.f, -c0, -c1, c0, 0.f};
    float* o6 = &sI6[36 * t];
#pragma unroll
    for (int r = 0; r < 3; ++r)
#pragma unroll
      for (int c = 0; c < 3; ++c) {
        const float cct = C[3 * r] * C[3 * c] + C[3 * r + 1] * C[3 * c + 1] + C[3 * r + 2] * C[3 * c + 2];
        o6[6 * r + c] = inertia[9 * t + 3 * r + c] + m * cct;  // I + m*cx*cx^T
        o6[6 * r + c + 3] = m * C[3 * r + c];                  // m*cx
        o6[6 * (r + 3) + c] = m * C[3 * c + r];                // m*cx^T
        o6[6 * (r + 3) + c + 3] = (r == c) ? m : 0.f;          // m*I
      }
  }
  __syncthreads();
  if (b >= B) return;

  // ---- load state -------------------------------------------------------
  float q[NJ], qd[NJ], tau[NJ];
#pragma unroll
  for (int j = 0; j < NJ; ++j) {
    q[j] = x[b * 14 + j];
    qd[j] = x[b * 14 + 7 + j];
    tau[j] = u[b * NJ + j];
  }

  float E[NJ][9], cc[NJ][6], pA[NJ][6];
  float wp0 = 0.f, wp1 = 0.f, wp2 = 0.f, up0 = 0.f, up1 = 0.f, up2 = 0.f;

  // ---- forward kinematics sweep ----------------------------------------
#pragma unroll
  for (int i = 0; i < NJ; ++i) {
    const float ax = sAxes[3 * i], ay = sAxes[3 * i + 1], az = sAxes[3 * i + 2];
    float sn, cn;
    sincosf(q[i], &sn, &cn);
    const float aa = ax * ax + ay * ay + az * az;
    const float oc = 1.f - cn;
    float R[9];  // rodrigues: I + s*K + (1-c)*K^2,  K^2 = a a^T - |a|^2 I
    R[0] = 1.f + oc * (ax * ax - aa);
    R[1] = -sn * az + oc * (ax * ay);
    R[2] = sn * ay + oc * (ax * az);
    R[3] = sn * az + oc * (ay * ax);
    R[4] = 1.f + oc * (ay * ay - aa);
    R[5] = -sn * ax + oc * (ay * az);
    R[6] = -sn * ay + oc * (az * ax);
    R[7] = sn * ax + oc * (az * ay);
    R[8] = 1.f + oc * (az * az - aa);
    const float* Rf = &sR[9 * i];
    float* Ei = E[i];
    // E = (Rfix * R)^T  :  E[r][c] = sum_k Rf[c][k] * R[k][r]
#pragma unroll
    for (int r = 0; r < 3; ++r)
#pragma unroll
      for (int c = 0; c < 3; ++c)
        Ei[3 * r + c] = Rf[3 * c + 0] * R[0 + r] + Rf[3 * c + 1] * R[3 + r] + Rf[3 * c + 2] * R[6 + r];

    const float qdi = qd[i];
    float w0, w1, w2, u0, u1, u2;
    if (i == 0) {
      w0 = qdi * ax; w1 = qdi * ay; w2 = qdi * az;
      u0 = 0.f; u1 = 0.f; u2 = 0.f;
    } else {
      const float px = sP[3 * i], py = sP[3 * i + 1], pz = sP[3 * i + 2];
      const float t0 = up0 - (py * wp2 - pz * wp1);  // u_prev - p x w_prev
      const float t1 = up1 - (pz * wp0 - px * wp2);
      const float t2 = up2 - (px * wp1 - py * wp0);
      w0 = Ei[0] * wp0 + Ei[1] * wp1 + Ei[2] * wp2 + qdi * ax;
      w1 = Ei[3] * wp0 + Ei[4] * wp1 + Ei[5] * wp2 + qdi * ay;
      w2 = Ei[6] * wp0 + Ei[7] * wp1 + Ei[8] * wp2 + qdi * az;
      u0 = Ei[0] * t0 + Ei[1] * t1 + Ei[2] * t2;
      u1 = Ei[3] * t0 + Ei[4] * t1 + Ei[5] * t2;
      u2 = Ei[6] * t0 + Ei[7] * t1 + Ei[8] * t2;
    }
    // c_i = qd_i * [ w x a , u x a ]
    cc[i][0] = qdi * (w1 * az - w2 * ay);
    cc[i][1] = qdi * (w2 * ax - w0 * az);
    cc[i][2] = qdi * (w0 * ay - w1 * ax);
    cc[i][3] = qdi * (u1 * az - u2 * ay);
    cc[i][4] = qdi * (u2 * ax - u0 * az);
    cc[i][5] = qdi * (u0 * ay - u1 * ax);
    // Iv = I6 * v   (LDS broadcast reads)
    const float* I6i = &sI6[36 * i];
    const float vv[6] = {w0, w1, w2, u0, u1, u2};
    float Iv[6];
#pragma unroll
    for (int k = 0; k < 6; ++k)
      Iv[k] = I6i[6 * k] * vv[0] + I6i[6 * k + 1] * vv[1] + I6i[6 * k + 2] * vv[2] +
              I6i[6 * k + 3] * vv[3] + I6i[6 * k + 4] * vv[4] + I6i[6 * k + 5] * vv[5];
    // pA = crf(v, Iv) = [ w x n + u x f , w x f ]
    pA[i][0] = (w1 * Iv[2] - w2 * Iv[1]) + (u1 * Iv[5] - u2 * Iv[4]);
    pA[i][1] = (w2 * Iv[0] - w0 * Iv[2]) + (u2 * Iv[3] - u0 * Iv[5]);
    pA[i][2] = (w0 * Iv[1] - w1 * Iv[0]) + (u0 * Iv[4] - u1 * Iv[3]);
    pA[i][3] = w1 * Iv[5] - w2 * Iv[4];
    pA[i][4] = w2 * Iv[3] - w0 * Iv[5];
    pA[i][5] = w0 * Iv[4] - w1 * Iv[3];
    wp0 = w0; wp1 = w1; wp2 = w2; up0 = u0; up1 = u1; up2 = u2;
  }

  // ---- backward articulated-inertia sweep ------------------------------
  float U[NJ][6], dinv[NJ], uu[NJ];
  float IAc[36];
  float pAc[6] = {0.f, 0.f, 0.f, 0.f, 0.f, 0.f};
#pragma unroll
  for (int k = 0; k < 36; ++k) IAc[k] = 0.f;

#pragma unroll
  for (int ii = 0; ii < NJ; ++ii) {
    const int i = NJ - 1 - ii;
    const float* I6i = &sI6[36 * i];
    float IA[36];
#pragma unroll
    for (int k = 0; k < 36; ++k) IA[k] = I6i[k] + IAc[k];
    float pAi[6];
#pragma unroll
    for (int k = 0; k < 6; ++k) pAi[k] = pA[i][k] + pAc[k];

    const float ax = sAxes[3 * i], ay = sAxes[3 * i + 1], az = sAxes[3 * i + 2];
    float Ui[6];
#pragma unroll
    for (int k = 0; k < 6; ++k)
      Ui[k] = IA[6 * k + 0] * ax + IA[6 * k + 1] * ay + IA[6 * k + 2] * az;
    const float d = Ui[0] * ax + Ui[1] * ay + Ui[2] * az;
    const float id = 1.f / d;
    const float uui = tau[i] - (pAi[0] * ax + pAi[1] * ay + pAi[2] * az);
#pragma unroll
    for (int k = 0; k < 6; ++k) U[i][k] = Ui[k];
    dinv[i] = id;
    uu[i] = uui;

    if (i > 0) {
      float Ia[36];
#pragma unroll
      for (int r = 0; r < 6; ++r)
#pragma unroll
        for (int c = 0; c < 6; ++c) Ia[6 * r + c] = IA[6 * r + c] - Ui[r] * Ui[c] * id;
      float pa[6];
      const float g = uui * id;
#pragma unroll
      for (int k = 0; k < 6; ++k)
        pa[k] = pAi[k] + Ui[k] * g +
                Ia[6 * k + 0] * cc[i][0] + Ia[6 * k + 1] * cc[i][1] + Ia[6 * k + 2] * cc[i][2] +
                Ia[6 * k + 3] * cc[i][3] + Ia[6 * k + 4] * cc[i][4] + Ia[6 * k + 5] * cc[i][5];

      const float* Eb = E[i];
      const float px = sP[3 * i], py = sP[3 * i + 1], pz = sP[3 * i + 2];
      float F[9];  // F = -E*skew(p):  row r = p x E_row_r
#pragma unroll
      for (int r = 0; r < 3; ++r) {
        F[3 * r + 0] = py * Eb[3 * r + 2] - pz * Eb[3 * r + 1];
        F[3 * r + 1] = pz * Eb[3 * r + 0] - px * Eb[3 * r + 2];
        F[3 * r + 2] = px * Eb[3 * r + 1] - py * Eb[3 * r + 0];
      }
      float Ab[9], Bb[9], Cb[9], Db[9];
#pragma unroll
      for (int r = 0; r < 3; ++r)
#pragma unroll
        for (int c = 0; c < 3; ++c) {
          Ab[3 * r + c] = Ia[6 * r + c];
          Bb[3 * r + c] = Ia[6 * r + c + 3];
          Cb[3 * r + c] = Ia[6 * (r + 3) + c];
          Db[3 * r + c] = Ia[6 * (r + 3) + c + 3];
        }
      // Y = Ia * Xm   with Xm = [[E,0],[F,E]]
      float Ytl[9], Ytr[9], Ybl[9], Ybr[9];
      m3mul(Ab, Eb, Ytl); m3mul_acc(Bb, F, Ytl);
      m3mul(Bb, Eb, Ytr);
      m3mul(Cb, Eb, Ybl); m3mul_acc(Db, F, Ybl);
      m3mul(Db, Eb, Ybr);
      // Z = Xm^T * Y  (symmetric; bottom-left = top-right^T)
      float Ztl[9], Ztr[9], Zbr[9];
      m3Tmul(Eb, Ytl, Ztl); m3Tmul_acc(F, Ybl, Ztl);
      m3Tmul(Eb, Ytr, Ztr); m3Tmul_acc(F, Ybr, Ztr);
      m3Tmul(Eb, Ybr, Zbr);
#pragma unroll
      for (int r = 0; r < 3; ++r)
#pragma unroll
        for (int c = 0; c < 3; ++c) {
          IAc[6 * r + c] = Ztl[3 * r + c];
          IAc[6 * r + c + 3] = Ztr[3 * r + c];
          IAc[6 * (r + 3) + c] = Ztr[3 * c + r];
          IAc[6 * (r + 3) + c + 3] = Zbr[3 * r + c];
        }
      // pA contribution to parent
      float fE[3], nT[3];
      m3Tvec(Eb, &pa[3], fE);
      m3Tvec(Eb, &pa[0], nT);
      pAc[0] = nT[0] + (py * fE[2] - pz * fE[1]);
      pAc[1] = nT[1] + (pz * fE[0] - px * fE[2]);
      pAc[2] = nT[2] + (px * fE[1] - py * fE[0]);
      pAc[3] = fE[0]; pAc[4] = fE[1]; pAc[5] = fE[2];
    }
  }

  // ---- forward acceleration sweep --------------------------------------
  float aprev[6] = {0.f, 0.f, 0.f, 0.f, 0.f, G_ACC};
  float qdd[NJ];
#pragma unroll
  for (int i = 0; i < NJ; ++i) {
    const float* Eb = E[i];
    const float px = sP[3 * i], py = sP[3 * i + 1], pz = sP[3 * i + 2];
    const float t0 = aprev[3] - (py * aprev[2] - pz * aprev[1]);
    const float t1 = aprev[4] - (pz * aprev[0] - px * aprev[2]);
    const float t2 = aprev[5] - (px * aprev[1] - py * aprev[0]);
    float ap[6];
    ap[0] = Eb[0] * aprev[0] + Eb[1] * aprev[1] + Eb[2] * aprev[2] + cc[i][0];
    ap[1] = Eb[3] * aprev[0] + Eb[4] * aprev[1] + Eb[5] * aprev[2] + cc[i][1];
    ap[2] = Eb[6] * aprev[0] + Eb[7] * aprev[1] + Eb[8] * aprev[2] + cc[i][2];
    ap[3] = Eb[0] * t0 + Eb[1] * t1 + Eb[2] * t2 + cc[i][3];
    ap[4] = Eb[3] * t0 + Eb[4] * t1 + Eb[5] * t2 + cc[i][4];
    ap[5] = Eb[6] * t0 + Eb[7] * t1 + Eb[8] * t2 + cc[i][5];
    const float num = uu[i] - (U[i][0] * ap[0] + U[i][1] * ap[1] + U[i][2] * ap[2] +
                               U[i][3] * ap[3] + U[i][4] * ap[4] + U[i][5] * ap[5]);
    const float qi = num * dinv[i];
    qdd[i] = qi;
    const float ax = sAxes[3 * i], ay = sAxes[3 * i + 1], az = sAxes[3 * i + 2];
    aprev[0] = ap[0] + qi * ax;
    aprev[1] = ap[1] + qi * ay;
    aprev[2] = ap[2] + qi * az;
    aprev[3] = ap[3]; aprev[4] = ap[4]; aprev[5] = ap[5];
  }

  // ---- semi-implicit Euler + store -------------------------------------
#pragma unroll
  for (int j = 0; j < NJ; ++j) {
    const float vn = qd[j] + DT_STEP * qdd[j];
    out[b * 14 + j] = q[j] + DT_STEP * vn;
    out[b * 14 + 7 + j] = vn;
  }
}

extern "C" void kernel_launch(void* const* d_in, const int* in_sizes, int n_in,
                              void* d_out, int out_size, void* d_ws, size_t ws_size,
                              hipStream_t stream) {
  const float* x = (const float*)d_in[0];
  const float* u = (const float*)d_in[1];
  const float* axes = (const float*)d_in[2];
  const float* Rfix = (const float*)d_in[3];
  const float* pfix = (const float*)d_in[4];
  const float* mass = (const float*)d_in[5];
  const float* com = (const float*)d_in[6];
  const float* inertia = (const float*)d_in[7];
  float* out = (float*)d_out;

  const int B = in_sizes[0] / 14;
  if (B <= 0) return;
  const int threads = 256;
  const int blocks = (B + threads - 1) / threads;
  hipLaunchKernelGGL(panda_aba_kernel, dim3(blocks), dim3(threads), 0, stream,
                     x, u, axes, Rfix, pfix, mass, com, inertia, out, B);
}